// HWFE_3015067042145
// MI455X (gfx1250) — compile-verified
//
#include <hip/hip_runtime.h>
#include <math.h>

// ---------------- constants ----------------
#define C0   128      // base channels
#define C4   512      // 4*C
#define H1   128      // half-res H
#define W1   128      // half-res W
#define HW1  16384    // H1*W1
#define XW   256      // full-res W
#define NB   8        // batch
#define EPSV 1e-5f

typedef float v2f __attribute__((ext_vector_type(2)));
typedef float v8f __attribute__((ext_vector_type(8)));

// interleaved LDS layout for B tiles: element (k, n) of a K x 16 tile
#define LIDX(k,n) ((((k)>>1)*16 + (n))*2 + ((k)&1))

__device__ __forceinline__ float gelu_f(float x) {
  // exact (erf-based) GELU to match jax.nn.gelu(approximate=False)
  return 0.5f * x * (1.0f + erff(x * 0.70710678118654752f));
}

// =====================================================================
// K1: Haar DWT + conv1x1 (Cin=512 -> Cout=128), WMMA f32 16x16x4
// one workgroup = 16 pixels x all 128 out channels (8 waves x 16 ch)
// =====================================================================
__global__ __launch_bounds__(256) void k1_dwt_conv1(
    const float* __restrict__ x, const float* __restrict__ w_c1,
    const float* __restrict__ b_c1, float* __restrict__ h_pre)
{
  __shared__ float bt[C4 * 16];          // 32 KB B-tile, interleaved
  const int t   = blockIdx.x;            // 8192 tiles
  const int b   = t >> 10;               // 1024 tiles per image
  const int rem = t & 1023;
  const int h2  = rem >> 3;
  const int wb  = (rem & 7) << 4;
  const int tid = threadIdx.x;

  // stage: DWT of x -> 512 x 16 tile in LDS
  for (int j = 0; j < 8; ++j) {
    int p  = tid + j * 256;              // 2048 (c,px) pairs
    int c  = p >> 4, px = p & 15;
    const float* xp = x + (((size_t)(b * C0 + c) * XW) + 2 * h2) * XW + 2 * (wb + px);
    float x00 = xp[0], x01 = xp[1], x10 = xp[XW], x11 = xp[XW + 1];
    bt[LIDX(c,        px)] = (x00 + x01 + x10 + x11) * 0.5f; // ll
    bt[LIDX(c + 128,  px)] = (x00 + x01 - x10 - x11) * 0.5f; // lh
    bt[LIDX(c + 256,  px)] = (x00 - x01 + x10 - x11) * 0.5f; // hl
    bt[LIDX(c + 384,  px)] = (x00 - x01 - x10 + x11) * 0.5f; // hh
  }
  __syncthreads();

  const int lane = tid & 31;
  const int wv   = tid >> 5;
  const int m    = lane & 15;            // M row / N col
  const int kh   = lane >> 4;            // K half
  const int ot   = wv * 16;              // out-channel tile base

  v8f acc = {};
  const float* wr = w_c1 + (size_t)(ot + m) * C4 + 2 * kh;
  for (int k0 = 0; k0 < C4; k0 += 4) {
    v2f a  = *(const v2f*)(wr + k0);
    v2f bf = *(const v2f*)&bt[(((k0 >> 1) + kh) * 16 + m) * 2];
    acc = __builtin_amdgcn_wmma_f32_16x16x4_f32(false, a, false, bf,
                                                (short)0, acc, false, false);
  }

  const int pix = h2 * W1 + wb + m;
  for (int r = 0; r < 8; ++r) {
    int ch = ot + r + kh * 8;
    h_pre[(size_t)(b * C0 + ch) * HW1 + pix] = acc[r] + b_c1[ch];
  }
}

// =====================================================================
// per-channel stats partials (for BN1 over h_pre, BN2 over pre2)
// grid = (64 slices, C); deterministic in-block tree reduction
// =====================================================================
__global__ __launch_bounds__(256) void k_chan_part(
    const float* __restrict__ src, float2* __restrict__ part, int C)
{
  __shared__ float s1[256], s2[256];
  const int c = blockIdx.y, s = blockIdx.x, tid = threadIdx.x;
  const int i = s * 256 + tid;           // 64*256 = 16384 = HW1
  float sum = 0.f, sq = 0.f;
  for (int b = 0; b < NB; ++b) {
    float v = src[(size_t)(b * C + c) * HW1 + i];
    sum += v; sq += v * v;
  }
  s1[tid] = sum; s2[tid] = sq; __syncthreads();
  for (int o = 128; o > 0; o >>= 1) {
    if (tid < o) { s1[tid] += s1[tid + o]; s2[tid] += s2[tid + o]; }
    __syncthreads();
  }
  if (tid == 0) { float2 r; r.x = s1[0]; r.y = s2[0]; part[c * 64 + s] = r; }
}

__global__ __launch_bounds__(512) void k_chan_fin(
    const float2* __restrict__ part, const float* __restrict__ g,
    const float* __restrict__ bv, float* __restrict__ ss, int C, float ninv)
{
  int c = threadIdx.x;
  if (c >= C) return;
  float sum = 0.f, sq = 0.f;
  for (int s = 0; s < 64; ++s) { float2 p = part[c * 64 + s]; sum += p.x; sq += p.y; }
  float mu  = sum * ninv;
  float var = sq * ninv - mu * mu;
  float rs  = rsqrtf(var + EPSV);
  float sc  = rs * g[c];
  ss[2 * c]     = sc;
  ss[2 * c + 1] = bv[c] - mu * sc;
}

// =====================================================================
// K4: depthwise 3x3 conv; BN1 affine + GELU fused on load
// =====================================================================
__global__ __launch_bounds__(256) void k4_dwconv(
    const float* __restrict__ h_pre, const float* __restrict__ bn1_ss,
    const float* __restrict__ dw_w, const float* __restrict__ dw_b,
    float* __restrict__ o_pre)
{
  int idx = blockIdx.x * 256 + threadIdx.x;           // 16.7M
  int w = idx & 127, h = (idx >> 7) & 127, c = (idx >> 14) & 127, b = idx >> 21;
  float sc = bn1_ss[2 * c], sh = bn1_ss[2 * c + 1];
  const float* wp  = dw_w + c * 9;
  const float* src = h_pre + (size_t)(b * C0 + c) * HW1;
  float acc = 0.f;
  for (int ky = -1; ky <= 1; ++ky) {
    int y = h + ky; if ((unsigned)y >= (unsigned)H1) continue;
    for (int kx = -1; kx <= 1; ++kx) {
      int xw = w + kx; if ((unsigned)xw >= (unsigned)W1) continue;
      float v = gelu_f(fmaf(src[y * W1 + xw], sc, sh));
      acc = fmaf(wp[(ky + 1) * 3 + (kx + 1)], v, acc);
    }
  }
  o_pre[idx] = acc + dw_b[c];
}

// =====================================================================
// per-sample layernorm stats (over C*H*W = 2,097,152 per sample)
// =====================================================================
__global__ __launch_bounds__(256) void k_ln_part(
    const float* __restrict__ src, float2* __restrict__ part)
{
  __shared__ float s1[256], s2[256];
  const int b = blockIdx.y, s = blockIdx.x, tid = threadIdx.x;
  const size_t base = (size_t)b * C0 * HW1 + (size_t)s * 4096;
  float sum = 0.f, sq = 0.f;
  for (int k = 0; k < 16; ++k) {
    float v = src[base + k * 256 + tid];
    sum += v; sq += v * v;
  }
  s1[tid] = sum; s2[tid] = sq; __syncthreads();
  for (int o = 128; o > 0; o >>= 1) {
    if (tid < o) { s1[tid] += s1[tid + o]; s2[tid] += s2[tid + o]; }
    __syncthreads();
  }
  if (tid == 0) { float2 r; r.x = s1[0]; r.y = s2[0]; part[b * 512 + s] = r; }
}

__global__ __launch_bounds__(64) void k_ln_fin(
    const float2* __restrict__ part, float* __restrict__ ms)
{
  int b = threadIdx.x;
  if (b >= NB) return;
  float sum = 0.f, sq = 0.f;
  for (int s = 0; s < 512; ++s) { float2 p = part[b * 512 + s]; sum += p.x; sq += p.y; }
  const float ninv = 1.0f / (float)(C0 * HW1);
  float mu  = sum * ninv;
  float var = sq * ninv - mu * mu;
  ms[2 * b]     = mu;
  ms[2 * b + 1] = rsqrtf(var + EPSV);
}

// =====================================================================
// K6: LN affine -> pw1 (WMMA) -> GELU -> pw2 (WMMA) -> + res -> h2
// h2 may alias o_pre: each WG reads its region fully (barrier) before writing
// =====================================================================
__global__ __launch_bounds__(256) void k6_mlp(
    const float* op, const float* __restrict__ h_pre,
    const float* __restrict__ ln_ms,
    const float* __restrict__ gn_g, const float* __restrict__ gn_b,
    const float* __restrict__ pw1_w, const float* __restrict__ pw1_b,
    const float* __restrict__ pw2_w, const float* __restrict__ pw2_b,
    const float* __restrict__ bn1_ss, float* h2)
{
  __shared__ float a_in[C0 * 16];
  __shared__ float a_mid[C0 * 16];
  const int t   = blockIdx.x;
  const int b   = t >> 10;
  const int rem = t & 1023;
  const int hh2 = rem >> 3;
  const int wb  = (rem & 7) << 4;
  const int tid = threadIdx.x;

  const float mu = ln_ms[2 * b], rsb = ln_ms[2 * b + 1];
  for (int j = 0; j < 8; ++j) {
    int p = tid + j * 256;               // 2048 = 128 x 16
    int c = p >> 4, px = p & 15;
    float v = op[(size_t)(b * C0 + c) * HW1 + hh2 * W1 + wb + px];
    a_in[LIDX(c, px)] = fmaf((v - mu) * rsb, gn_g[c], gn_b[c]);
  }
  __syncthreads();

  const int lane = tid & 31;
  const int wv   = tid >> 5;
  const int m    = lane & 15;
  const int kh   = lane >> 4;
  const int ot   = wv * 16;

  // pw1
  v8f acc = {};
  {
    const float* wr = pw1_w + (size_t)(ot + m) * C0 + 2 * kh;
    for (int k0 = 0; k0 < C0; k0 += 4) {
      v2f a  = *(const v2f*)(wr + k0);
      v2f bf = *(const v2f*)&a_in[(((k0 >> 1) + kh) * 16 + m) * 2];
      acc = __builtin_amdgcn_wmma_f32_16x16x4_f32(false, a, false, bf,
                                                  (short)0, acc, false, false);
    }
  }
  for (int r = 0; r < 8; ++r) {
    int ch = ot + r + kh * 8;
    a_mid[LIDX(ch, m)] = gelu_f(acc[r] + pw1_b[ch]);
  }
  __syncthreads();

  // pw2
  v8f acc2 = {};
  {
    const float* wr = pw2_w + (size_t)(ot + m) * C0 + 2 * kh;
    for (int k0 = 0; k0 < C0; k0 += 4) {
      v2f a  = *(const v2f*)(wr + k0);
      v2f bf = *(const v2f*)&a_mid[(((k0 >> 1) + kh) * 16 + m) * 2];
      acc2 = __builtin_amdgcn_wmma_f32_16x16x4_f32(false, a, false, bf,
                                                   (short)0, acc2, false, false);
    }
  }
  const int pix = hh2 * W1 + wb + m;
  for (int r = 0; r < 8; ++r) {
    int ch = ot + r + kh * 8;
    float res = gelu_f(fmaf(h_pre[(size_t)(b * C0 + ch) * HW1 + pix],
                            bn1_ss[2 * ch], bn1_ss[2 * ch + 1]));
    h2[(size_t)(b * C0 + ch) * HW1 + pix] = acc2[r] + pw2_b[ch] + res;
  }
}

// =====================================================================
// K7: conv1x1 Cin=128 -> Cout=512; each wave does 4 WMMA out tiles
// =====================================================================
__global__ __launch_bounds__(256) void k7_conv2(
    const float* __restrict__ h2, const float* __restrict__ w_c2,
    const float* __restrict__ b_c2, float* __restrict__ pre2)
{
  __shared__ float a_in[C0 * 16];
  const int t   = blockIdx.x;
  const int b   = t >> 10;
  const int rem = t & 1023;
  const int hh2 = rem >> 3;
  const int wb  = (rem & 7) << 4;
  const int tid = threadIdx.x;

  for (int j = 0; j < 8; ++j) {
    int p = tid + j * 256;
    int c = p >> 4, px = p & 15;
    a_in[LIDX(c, px)] = h2[(size_t)(b * C0 + c) * HW1 + hh2 * W1 + wb + px];
  }
  __syncthreads();

  const int lane = tid & 31;
  const int wv   = tid >> 5;
  const int m    = lane & 15;
  const int kh   = lane >> 4;
  const int pix  = hh2 * W1 + wb + m;

  for (int tb = 0; tb < 4; ++tb) {
    const int ot = wv * 16 + tb * 128;   // covers 0..511
    v8f acc = {};
    const float* wr = w_c2 + (size_t)(ot + m) * C0 + 2 * kh;
    for (int k0 = 0; k0 < C0; k0 += 4) {
      v2f a  = *(const v2f*)(wr + k0);
      v2f bf = *(const v2f*)&a_in[(((k0 >> 1) + kh) * 16 + m) * 2];
      acc = __builtin_amdgcn_wmma_f32_16x16x4_f32(false, a, false, bf,
                                                  (short)0, acc, false, false);
    }
    for (int r = 0; r < 8; ++r) {
      int ch = ot + r + kh * 8;
      pre2[(size_t)(b * C4 + ch) * HW1 + pix] = acc[r] + b_c2[ch];
    }
  }
}

// =====================================================================
// K9: BN2 affine + GELU + Haar IDWT -> output (8,128,256,256)
// =====================================================================
__global__ __launch_bounds__(256) void k9_idwt(
    const float* __restrict__ pre2, const float* __restrict__ bn2_ss,
    float* __restrict__ out)
{
  int idx = blockIdx.x * 256 + threadIdx.x;   // 16.7M over (b,c,h2,w2)
  int w2 = idx & 127, h2 = (idx >> 7) & 127, c = (idx >> 14) & 127, b = idx >> 21;
  int pix = h2 * W1 + w2;
  float v[4];
  for (int s = 0; s < 4; ++s) {
    int ch = c + s * 128;
    float raw = pre2[(size_t)(b * C4 + ch) * HW1 + pix];
    v[s] = gelu_f(fmaf(raw, bn2_ss[2 * ch], bn2_ss[2 * ch + 1]));
  }
  float ll = v[0], lh = v[1], hl = v[2], hh = v[3];
  float y00 = (ll + lh + hl + hh) * 0.5f;
  float y01 = (ll + lh - hl - hh) * 0.5f;
  float y10 = (ll - lh + hl - hh) * 0.5f;
  float y11 = (ll - lh - hl + hh) * 0.5f;
  float* opx = out + (((size_t)(b * C0 + c) * XW) + 2 * h2) * XW + 2 * w2;
  opx[0]  = y00; opx[1]      = y01;
  opx[XW] = y10; opx[XW + 1] = y11;
}

// =====================================================================
extern "C" void kernel_launch(void* const* d_in, const int* in_sizes, int n_in,
                              void* d_out, int out_size, void* d_ws, size_t ws_size,
                              hipStream_t stream) {
  const float* x     = (const float*)d_in[0];
  const float* w_c1  = (const float*)d_in[1];
  const float* b_c1  = (const float*)d_in[2];
  const float* bn1_g = (const float*)d_in[3];
  const float* bn1_b = (const float*)d_in[4];
  const float* dw_w  = (const float*)d_in[5];
  const float* dw_b  = (const float*)d_in[6];
  const float* gn_g  = (const float*)d_in[7];
  const float* gn_b  = (const float*)d_in[8];
  const float* pw1_w = (const float*)d_in[9];
  const float* pw1_b = (const float*)d_in[10];
  const float* pw2_w = (const float*)d_in[11];
  const float* pw2_b = (const float*)d_in[12];
  const float* w_c2  = (const float*)d_in[13];
  const float* b_c2  = (const float*)d_in[14];
  const float* bn2_g = (const float*)d_in[15];
  const float* bn2_b = (const float*)d_in[16];

  // workspace layout (floats)
  float* ws = (float*)d_ws;
  const size_t n_half = (size_t)NB * C0 * HW1;      // 16,777,216
  float*  h_pre   = ws;                             // conv1 pre-activation
  float*  o_pre   = h_pre + n_half;                 // dwconv out; later aliased as h2
  float*  pre2    = o_pre + n_half;                 // conv2 pre-activation (4x size)
  float*  tail    = pre2 + (size_t)NB * C4 * HW1;
  float2* bn1_pt  = (float2*)tail;                  // 128*64
  float*  bn1_ss  = tail + 128 * 64 * 2;            // 128*2 (scale,shift)
  float2* ln_pt   = (float2*)(bn1_ss + 256);        // 8*512
  float*  ln_ms   = bn1_ss + 256 + 8 * 512 * 2;     // 8*2 (mu,rs)
  float2* bn2_pt  = (float2*)(ln_ms + 16);          // 512*64
  float*  bn2_ss  = ln_ms + 16 + 512 * 64 * 2;      // 512*2

  const float ninv_bn = 1.0f / (float)(NB * HW1);   // 1/131072

  // 1) DWT + conv1 (WMMA)
  k1_dwt_conv1<<<8192, 256, 0, stream>>>(x, w_c1, b_c1, h_pre);
  // 2) BN1 stats
  k_chan_part<<<dim3(64, 128), 256, 0, stream>>>(h_pre, bn1_pt, C0);
  k_chan_fin<<<1, 512, 0, stream>>>(bn1_pt, bn1_g, bn1_b, bn1_ss, C0, ninv_bn);
  // 3) dwconv3x3 with fused BN1+GELU
  k4_dwconv<<<65536, 256, 0, stream>>>(h_pre, bn1_ss, dw_w, dw_b, o_pre);
  // 4) LN stats
  k_ln_part<<<dim3(512, 8), 256, 0, stream>>>(o_pre, ln_pt);
  k_ln_fin<<<1, 64, 0, stream>>>(ln_pt, ln_ms);
  // 5) LN -> pw1 -> GELU -> pw2 -> +res  (h2 aliases o_pre; safe per-WG)
  k6_mlp<<<8192, 256, 0, stream>>>(o_pre, h_pre, ln_ms, gn_g, gn_b,
                                   pw1_w, pw1_b, pw2_w, pw2_b, bn1_ss, o_pre);
  // 6) conv2 (WMMA, 128->512)
  k7_conv2<<<8192, 256, 0, stream>>>(o_pre, w_c2, b_c2, pre2);
  // 7) BN2 stats
  k_chan_part<<<dim3(64, 512), 256, 0, stream>>>(pre2, bn2_pt, C4);
  k_chan_fin<<<1, 512, 0, stream>>>(bn2_pt, bn2_g, bn2_b, bn2_ss, C4, ninv_bn);
  // 8) BN2 + GELU + IDWT
  k9_idwt<<<65536, 256, 0, stream>>>(pre2, bn2_ss, (float*)d_out);
}